// GripNetExternalModule_66340064854088
// MI455X (gfx1250) — compile-verified
//
#include <hip/hip_runtime.h>

typedef __attribute__((ext_vector_type(2))) float v2f;
typedef __attribute__((ext_vector_type(8))) float v8f;

#define N_SRC  100000
#define N_OUT  50000
#define N_EDGE 500000
#define D      128

// ---------------------------------------------------------------------------
// init: out = 0, deg = 1.0 (self-loop fill weight)
// ---------------------------------------------------------------------------
__global__ void init_kernel(float* __restrict__ out, float* __restrict__ deg) {
    int i = blockIdx.x * blockDim.x + threadIdx.x;
    int stride = gridDim.x * blockDim.x;
    for (int t = i; t < N_OUT * D; t += stride) out[t] = 0.0f;
    for (int t = i; t < N_OUT; t += stride) deg[t] = 1.0f;
}

// ---------------------------------------------------------------------------
// deg[r] += w_e  scattered over edge rows
// ---------------------------------------------------------------------------
__global__ void deg_kernel(const int* __restrict__ row,
                           const float* __restrict__ w,
                           float* __restrict__ deg) {
    int i = blockIdx.x * blockDim.x + threadIdx.x;
    int stride = gridDim.x * blockDim.x;
    for (int e = i; e < N_EDGE; e += stride) {
        atomicAdd(&deg[row[e]], w[e]);
    }
}

// ---------------------------------------------------------------------------
// deg -> rsqrt(deg)  (deg >= 1 always)
// ---------------------------------------------------------------------------
__global__ void dinv_kernel(float* __restrict__ deg) {
    int i = blockIdx.x * blockDim.x + threadIdx.x;
    if (i < N_OUT) deg[i] = rsqrtf(deg[i]);
}

// ---------------------------------------------------------------------------
// y = diag(dinv) * (x[0:N_OUT] @ W)  via V_WMMA_F32_16X16X4_F32
// block = 128 threads (4 waves); block -> one 16-row stripe x 128 cols
// wave w -> column tiles [32w,32w+16) and [32w+16,32w+32)
// ---------------------------------------------------------------------------
__global__ __launch_bounds__(128) void gemm_kernel(const float* __restrict__ x,
                                                   const float* __restrict__ W,
                                                   const float* __restrict__ dinv,
                                                   float* __restrict__ y) {
    const int mt   = blockIdx.x;          // 0..3124
    const int wave = threadIdx.x >> 5;    // 0..3
    const int lane = threadIdx.x & 31;
    const int m    = lane & 15;           // A row / B col / D col within tile
    const int kh   = lane >> 4;           // K half-select
    const int row0 = mt * 16;
    const int col0 = wave * 32;

    v8f c0 = {};
    v8f c1 = {};

    const float* xrow = x + (size_t)(row0 + m) * D + 2 * kh;
    for (int k = 0; k < D; k += 4) {
        // A fragment: lane holds x[row0+m][k+2kh .. k+2kh+1] (8B contiguous load)
        v2f a = *(const v2f*)(xrow + k);
        const int kk = k + 2 * kh;
        const float* wb = W + (size_t)kk * D;
        // B fragments: lane holds W[kk][col] and W[kk+1][col]
        v2f b0 = { wb[col0 + m],      wb[D + col0 + m]      };
        v2f b1 = { wb[col0 + 16 + m], wb[D + col0 + 16 + m] };
        c0 = __builtin_amdgcn_wmma_f32_16x16x4_f32(false, a, false, b0,
                                                   (short)0, c0, false, false);
        c1 = __builtin_amdgcn_wmma_f32_16x16x4_f32(false, a, false, b1,
                                                   (short)0, c1, false, false);
    }

    // epilogue: scale row by dinv, store. D element (v): M = v + 8*kh, N = m
    #pragma unroll
    for (int v = 0; v < 8; ++v) {
        const int mrow = row0 + v + 8 * kh;
        const float dv = dinv[mrow];
        y[(size_t)mrow * D + col0 + m]      = c0[v] * dv;
        y[(size_t)mrow * D + col0 + 16 + m] = c1[v] * dv;
    }
}

// ---------------------------------------------------------------------------
// edge scatter: one wave per edge, float4 gather of y[row], 4 f32 atomics/lane
// y and out both fit in the 192MB L2 -> atomics resolve at L2
// ---------------------------------------------------------------------------
__global__ void scatter_kernel(const int* __restrict__ row,
                               const int* __restrict__ col,
                               const float* __restrict__ ew,
                               const float* __restrict__ y,
                               float* __restrict__ out) {
    int tid    = blockIdx.x * blockDim.x + threadIdx.x;
    int lane   = tid & 31;
    int waveId = tid >> 5;
    int nwaves = (gridDim.x * blockDim.x) >> 5;
    for (int e = waveId; e < N_EDGE; e += nwaves) {
        const int   r = row[e];
        const int   d = col[e];
        const float w = ew[e];
        const float4 v = ((const float4*)(y + (size_t)r * D))[lane];
        float* dst = out + (size_t)d * D + lane * 4;
        atomicAdd(dst + 0, v.x * w);
        atomicAdd(dst + 1, v.y * w);
        atomicAdd(dst + 2, v.z * w);
        atomicAdd(dst + 3, v.w * w);
    }
}

// ---------------------------------------------------------------------------
// out = relu(out + b), vectorized float4
// ---------------------------------------------------------------------------
__global__ void finalize_kernel(float* __restrict__ out, const float* __restrict__ b) {
    int i = blockIdx.x * blockDim.x + threadIdx.x;
    int stride = gridDim.x * blockDim.x;
    const int total4 = N_OUT * D / 4;
    const float4* b4 = (const float4*)b;
    float4* o4 = (float4*)out;
    for (int t = i; t < total4; t += stride) {
        float4 bb = b4[t & 31];   // 128 floats = 32 float4 per row
        float4 v  = o4[t];
        v.x = fmaxf(v.x + bb.x, 0.0f);
        v.y = fmaxf(v.y + bb.y, 0.0f);
        v.z = fmaxf(v.z + bb.z, 0.0f);
        v.w = fmaxf(v.w + bb.w, 0.0f);
        o4[t] = v;
    }
}

extern "C" void kernel_launch(void* const* d_in, const int* in_sizes, int n_in,
                              void* d_out, int out_size, void* d_ws, size_t ws_size,
                              hipStream_t stream) {
    const float* x  = (const float*)d_in[0];
    const int*   ei = (const int*)d_in[1];     // [2, N_EDGE] flat
    const float* ew = (const float*)d_in[2];
    const float* W  = (const float*)d_in[3];
    const float* b  = (const float*)d_in[4];
    float* out = (float*)d_out;

    // workspace: deg/dinv [N_OUT] then y [N_OUT x D]  (~25.8 MB total)
    float* deg = (float*)d_ws;
    float* y   = deg + N_OUT;

    const int* row = ei;            // destination-degree side in reference norm
    const int* col = ei + N_EDGE;   // output node index (already 0..N_OUT)

    init_kernel<<<512, 256, 0, stream>>>(out, deg);
    deg_kernel<<<(N_EDGE + 255) / 256, 256, 0, stream>>>(row, ew, deg);
    dinv_kernel<<<(N_OUT + 255) / 256, 256, 0, stream>>>(deg);
    gemm_kernel<<<N_OUT / 16, 128, 0, stream>>>(x, W, deg, y);
    scatter_kernel<<<1024, 256, 0, stream>>>(row, col, ew, y, out);
    finalize_kernel<<<512, 256, 0, stream>>>(out, b);
}